// CrossSessionCenterAlignMarginLoss_1537598292251
// MI455X (gfx1250) — compile-verified
//
#include <hip/hip_runtime.h>
#include <math.h>

// CrossSessionCenterAlignMarginLoss for MI455X (gfx1250, wave32).
// Bandwidth-bound: 2 passes x 128MB over features ~ 256MB @ 23.3TB/s ~ 11us floor
// (pass 2 should mostly hit the 192MB L2). The 4x2048 center Gram (all pairwise
// center cosines + norms) is computed with V_WMMA_F32_16X16X4_F32, K-split over
// 8 waves (64 WMMAs each) with uniform (non-divergent) operand loads.
// Fully deterministic: no float atomics, fixed-order two-stage reductions.

typedef __attribute__((ext_vector_type(2))) float v2f;
typedef __attribute__((ext_vector_type(8))) float v8f;

#define B_ROWS   16384
#define D_COLS   2048
#define NGRP     4

#define NPART    128            // k_segsum blocks (partial-sum buffers)
#define SEG_T    512            // threads per segsum block (16 waves)
#define RL_BLOCKS 256           // k_rowloss blocks
#define RL_T     256            // threads per rowloss block (8 waves)
#define RL_NWAVES (RL_BLOCKS * (RL_T / 32))   // 2048

// Workspace layout (float offsets); total ~1,061,392 floats (~4.25 MB)
#define WS_PART     0                         // NPART * 4 * 2048 = 1048576
#define WS_CPART    1048576                   // NPART * 4 = 512
#define WS_CENTERS  1049088                   // 4 * 2048 = 8192
#define WS_ZBUF     1057280                   // 2048 (must be zero; k_zero)
#define WS_GRAM     1059328                   // 16
#define WS_ROWPART  1059344                   // RL_NWAVES = 2048

#define EPSF 1e-8f

__global__ void k_zero(float* __restrict__ zbuf) {
    int i = blockIdx.x * blockDim.x + threadIdx.x;
    if (i < D_COLS) zbuf[i] = 0.0f;
}

// 128 blocks x 512 threads; block owns 128 consecutive rows, thread t owns
// columns [t*4, t*4+4). g is wave-uniform per row -> switch with static
// register indexing (no scratch). Per-block partial sums, no atomics.
__global__ void k_segsum(const float* __restrict__ feats,
                         const int*   __restrict__ labels,
                         const int*   __restrict__ sess,
                         float* __restrict__ part,
                         float* __restrict__ cpart) {
    const int tid        = threadIdx.x;          // 0..511
    const int colBase    = tid * 4;
    const int rowsPerBlk = B_ROWS / NPART;       // 128
    const int row0       = blockIdx.x * rowsPerBlk;

    float4 acc[4];
    int    cnt[4] = {0, 0, 0, 0};
#pragma unroll
    for (int i = 0; i < 4; ++i) acc[i] = make_float4(0.f, 0.f, 0.f, 0.f);

    for (int r = 0; r < rowsPerBlk; ++r) {
        const int row = row0 + r;
        const int g   = labels[row] * 2 + sess[row];
        const float4 x = *(const float4*)(feats + (size_t)row * D_COLS + colBase);
        if (r + 1 < rowsPerBlk) {
            __builtin_prefetch(feats + (size_t)(row + 1) * D_COLS + colBase, 0, 1);
        }
#define ACCUM(i) acc[i].x += x.x; acc[i].y += x.y; acc[i].z += x.z; acc[i].w += x.w; cnt[i]++
        switch (g) {
            case 0:  ACCUM(0); break;
            case 1:  ACCUM(1); break;
            case 2:  ACCUM(2); break;
            default: ACCUM(3); break;
        }
#undef ACCUM
    }

    float* p = part + (size_t)blockIdx.x * (NGRP * D_COLS);
#pragma unroll
    for (int gi = 0; gi < 4; ++gi) {
        *(float4*)(p + gi * D_COLS + colBase) = acc[gi];
    }
    if (tid == 0) {
#pragma unroll
        for (int gi = 0; gi < 4; ++gi) cpart[blockIdx.x * 4 + gi] = (float)cnt[gi];
    }
}

// Fixed-order reduction of per-block partials -> centers. Deterministic.
__global__ void k_reduce_centers(const float* __restrict__ part,
                                 const float* __restrict__ cpart,
                                 float* __restrict__ centers) {
    const int c = blockIdx.x * blockDim.x + threadIdx.x;   // 0..8191
    if (c >= NGRP * D_COLS) return;
    float s = 0.0f;
    for (int b = 0; b < NPART; ++b) s += part[(size_t)b * (NGRP * D_COLS) + c];
    const int g = c / D_COLS;
    float cnt = 0.0f;
    for (int b = 0; b < NPART; ++b) cnt += cpart[b * 4 + g];
    centers[c] = s / cnt;
}

// Gram = C * C^T, C is 4x2048 padded to 16 rows via a zeroed scratch row.
// 8 waves, each owns a 256-column K-slice (64 x V_WMMA_F32_16X16X4_F32),
// partials reduced via LDS. ISA layouts: A(16x4): lane L<16 holds
// {A[L][k],A[L][k+1]}, lane L+16 holds {A[L][k+2],A[L][k+3]}; B(4x16)=C^T has
// identical per-lane contents, so one v2f feeds both operands. Loads are
// unconditional (no EXEC divergence) -> pipelined. D: acc[v]=D[v][lane], lane<16.
__global__ void k_gram(const float* __restrict__ centers,
                       const float* __restrict__ zbuf,
                       float* __restrict__ gram) {
    __shared__ float sPart[8 * 16];
    const int tid  = threadIdx.x;        // 0..255
    const int w    = tid >> 5;           // wave id, uniform per wave
    const int lane = tid & 31;
    const int row  = lane & 15;
    const int koff = (lane >> 4) * 2;
    const float* base = (row < NGRP) ? (centers + row * D_COLS) : zbuf;
    const int kSlice = D_COLS / 8;       // 256
    const int k0 = w * kSlice;

    v8f acc = {};
    for (int k = k0; k < k0 + kSlice; k += 4) {
        v2f ab;
        ab.x = base[k + koff];
        ab.y = base[k + koff + 1];
        acc = __builtin_amdgcn_wmma_f32_16x16x4_f32(
            /*neg_a=*/false, ab, /*neg_b=*/false, ab,
            /*c_mod=*/(short)0, acc, /*reuse_a=*/false, /*reuse_b=*/false);
    }
    if (lane < NGRP) {
#pragma unroll
        for (int i = 0; i < 4; ++i) sPart[w * 16 + i * 4 + lane] = acc[i];
    }
    __syncthreads();
    if (tid < 16) {
        float s = 0.0f;
#pragma unroll
        for (int w2 = 0; w2 < 8; ++w2) s += sPart[w2 * 16 + tid];
        gram[tid] = s;
    }
}

// One wave per row (grid-strided); centers staged in 32KB LDS; wave32
// xor-reduction; one deterministic partial per wave (no atomics).
__global__ void k_rowloss(const float* __restrict__ feats,
                          const int*   __restrict__ labels,
                          const int*   __restrict__ sess,
                          const float* __restrict__ centers,
                          const float* __restrict__ gram,
                          float* __restrict__ rowpart) {
    __shared__ float sC[NGRP * D_COLS];
    for (int i = threadIdx.x; i < NGRP * D_COLS; i += blockDim.x) sC[i] = centers[i];
    __syncthreads();

    const int wpb  = blockDim.x >> 5;
    const int wave = blockIdx.x * wpb + (threadIdx.x >> 5);
    const int lane = threadIdx.x & 31;
    const int nW   = gridDim.x * wpb;

    float acc = 0.0f;
    for (int row = wave; row < B_ROWS; row += nW) {
        const int g = labels[row] * 2 + sess[row];
        const float* f = feats + (size_t)row * D_COLS;
        const float* c = sC + g * D_COLS;
        float dfc = 0.0f, dff = 0.0f;
        for (int k = lane * 4; k < D_COLS; k += 32 * 4) {
            const float4 fv = *(const float4*)(f + k);
            const float4 cv = *(const float4*)(c + k);
            dfc += fv.x * cv.x + fv.y * cv.y + fv.z * cv.z + fv.w * cv.w;
            dff += fv.x * fv.x + fv.y * fv.y + fv.z * fv.z + fv.w * fv.w;
        }
#pragma unroll
        for (int off = 16; off > 0; off >>= 1) {
            dfc += __shfl_xor(dfc, off, 32);
            dff += __shfl_xor(dff, off, 32);
        }
        if (lane == 0) {
            const float fn  = sqrtf(dff);
            const float cn  = sqrtf(gram[g * NGRP + g]);
            const float den = fmaxf(fn * cn, EPSF);
            acc += 1.0f - dfc / den;
        }
    }
    if (lane == 0) rowpart[wave] = acc;
}

// Fixed-order tree reduction of wave partials + final scalar math.
__global__ void k_finalize(const float* __restrict__ gram,
                           const float* __restrict__ rowpart,
                           float* __restrict__ out) {
    __shared__ float sred[256];
    float s = 0.0f;
    for (int i = threadIdx.x; i < RL_NWAVES; i += 256) s += rowpart[i];
    sred[threadIdx.x] = s;
    __syncthreads();
#pragma unroll
    for (int off = 128; off > 0; off >>= 1) {
        if (threadIdx.x < off) sred[threadIdx.x] += sred[threadIdx.x + off];
        __syncthreads();
    }
    if (threadIdx.x == 0) {
        float cn[4];
#pragma unroll
        for (int i = 0; i < 4; ++i) cn[i] = sqrtf(gram[i * NGRP + i]);
        const float cos01 = gram[0 * NGRP + 1] / fmaxf(cn[0] * cn[1], EPSF);
        const float cos23 = gram[2 * NGRP + 3] / fmaxf(cn[2] * cn[3], EPSF);
        const float cos02 = gram[0 * NGRP + 2] / fmaxf(cn[0] * cn[2], EPSF);
        const float cos03 = gram[0 * NGRP + 3] / fmaxf(cn[0] * cn[3], EPSF);
        const float cos12 = gram[1 * NGRP + 2] / fmaxf(cn[1] * cn[2], EPSF);
        const float cos13 = gram[1 * NGRP + 3] / fmaxf(cn[1] * cn[3], EPSF);

        const float center_loss = sred[0] / (float)B_ROWS;
        const float align_loss  = ((1.0f - cos01) + (1.0f - cos23)) * 0.5f;
        const float margin_loss = (cos02 + cos03 + cos12 + cos13) * 0.25f;
        const float total = 1.0f * center_loss + 0.1f * align_loss + 0.05f * margin_loss;
        out[0] = total;
        out[1] = center_loss;
        out[2] = align_loss;
        out[3] = margin_loss;
    }
}

extern "C" void kernel_launch(void* const* d_in, const int* in_sizes, int n_in,
                              void* d_out, int out_size, void* d_ws, size_t ws_size,
                              hipStream_t stream) {
    const float* feats  = (const float*)d_in[0];
    const int*   labels = (const int*)d_in[1];
    const int*   sess   = (const int*)d_in[2];
    float* ws  = (float*)d_ws;
    float* out = (float*)d_out;

    float* part    = ws + WS_PART;
    float* cpart   = ws + WS_CPART;
    float* centers = ws + WS_CENTERS;
    float* zbuf    = ws + WS_ZBUF;
    float* gram    = ws + WS_GRAM;
    float* rowpart = ws + WS_ROWPART;

    k_zero<<<(D_COLS + 255) / 256, 256, 0, stream>>>(zbuf);
    k_segsum<<<NPART, SEG_T, 0, stream>>>(feats, labels, sess, part, cpart);
    k_reduce_centers<<<(NGRP * D_COLS + 255) / 256, 256, 0, stream>>>(part, cpart, centers);
    k_gram<<<1, 256, 0, stream>>>(centers, zbuf, gram);
    k_rowloss<<<RL_BLOCKS, RL_T, 0, stream>>>(feats, labels, sess, centers, gram, rowpart);
    k_finalize<<<1, 256, 0, stream>>>(gram, rowpart, out);
}